// Criterion_64965675319881
// MI455X (gfx1250) — compile-verified
//
#include <hip/hip_runtime.h>
#include <hip/hip_bf16.h>
#include <math.h>

// ---------------------------------------------------------------------------
// MI455X (gfx1250) fused criterion:
//   loss_rank : 2048x16384x512 fp32 GEMM (batch @ norm(class_map)^T / T)
//               fused with online row log-softmax + label gather
//   loss_kd   : dual Gram matrices (batch@batch^T, teacher@teacher^T) fused
//               with masked-scale row log-softmax + KL reduction
// All matrix math on V_WMMA_F32_16X16X4_F32 (fp32 tensor pipe) to match the
// fp32 reference numerics. Whole working set (~40MB) fits in the 192MB L2.
// ---------------------------------------------------------------------------

typedef __attribute__((ext_vector_type(2))) float v2f;
typedef __attribute__((ext_vector_type(8))) float v8f;

#define B_ROWS   2048
#define E_DIM    512
#define C_DIM    16384
#define LDSSTR   516          // 512 + 4 pad: rows land on distinct LDS banks

__device__ __forceinline__ v8f wmma4(v2f a, v2f b, v8f c) {
  // D = A(16x4 f32) x B(4x16 f32) + C(16x16 f32)
  return __builtin_amdgcn_wmma_f32_16x16x4_f32(
      false, a, false, b, (short)0, c, false, false);
}

// Butterfly reductions inside each 16-lane half of a wave32 (WMMA row group)
__device__ __forceinline__ float half16_max(float v) {
#pragma unroll
  for (int m = 8; m >= 1; m >>= 1) v = fmaxf(v, __shfl_xor(v, m, 16));
  return v;
}
__device__ __forceinline__ float half16_sum(float v) {
#pragma unroll
  for (int m = 8; m >= 1; m >>= 1) v += __shfl_xor(v, m, 16);
  return v;
}

// ---------------------------------------------------------------------------
// K0: zero the two scalar accumulators (ws is poisoned by harness)
__global__ void init_kernel(float* accs) {
  if (threadIdx.x < 2) accs[threadIdx.x] = 0.0f;
}

// K1: rnorm[c] = 1/||class_map[c]||   (one wave per row)
__global__ __launch_bounds__(256) void rnorm_kernel(const float* __restrict__ cm,
                                                    float* __restrict__ rnorm) {
  int wave = threadIdx.x >> 5, lane = threadIdx.x & 31;
  int row = blockIdx.x * 8 + wave;
  const float* p = cm + (size_t)row * E_DIM;
  float s = 0.0f;
#pragma unroll 4
  for (int e = lane; e < E_DIM; e += 32) { float v = p[e]; s += v * v; }
#pragma unroll
  for (int m = 16; m >= 1; m >>= 1) s += __shfl_xor(s, m, 32);
  if (lane == 0) rnorm[row] = rsqrtf(s);
}

// K2: label_logit[b] = dot(batch[b], class_map[labels[b]]) * rnorm[lab] / T
__global__ __launch_bounds__(256) void label_logit_kernel(
    const float* __restrict__ batch, const float* __restrict__ cm,
    const int* __restrict__ labels, const float* __restrict__ rnorm,
    float* __restrict__ out) {
  int wave = threadIdx.x >> 5, lane = threadIdx.x & 31;
  int row = blockIdx.x * 8 + wave;
  int lab = labels[row];
  const float* bp = batch + (size_t)row * E_DIM;
  const float* wp = cm + (size_t)lab * E_DIM;
  float s = 0.0f;
#pragma unroll 4
  for (int e = lane; e < E_DIM; e += 32) s += bp[e] * wp[e];
#pragma unroll
  for (int m = 16; m >= 1; m >>= 1) s += __shfl_xor(s, m, 32);
  if (lane == 0) out[row] = s * rnorm[lab] * (1.0f / 0.05f);
}

// K3: fused logits GEMM + online row (max, sumexp) partials.
// grid = (B/16 rowblocks, 8 column chunks of 2048); 8 waves/WG, wave = one
// 16x16 WMMA tile, looping over 16 tiles of its chunk. A-block staged in LDS.
__global__ __launch_bounds__(256) void logits_stats_kernel(
    const float* __restrict__ batch, const float* __restrict__ cm,
    const float* __restrict__ rnorm, float* __restrict__ part) {
  __shared__ float As[16 * LDSSTR];
  const int tid = threadIdx.x;
  const int wave = tid >> 5, lane = tid & 31;
  const int rowBase = blockIdx.x * 16;
  const int chunk = blockIdx.y;                 // 2048 cols per chunk

  for (int i = tid; i < 16 * (E_DIM / 4); i += 256) {
    int r = i / (E_DIM / 4), c4 = (i % (E_DIM / 4)) * 4;
    float4 v = *(const float4*)(batch + (size_t)(rowBase + r) * E_DIM + c4);
    float* d = &As[r * LDSSTR + c4];
    d[0] = v.x; d[1] = v.y; d[2] = v.z; d[3] = v.w;
  }
  __syncthreads();

  const int half = lane >> 4, l16 = lane & 15;
  const float* aBase = &As[l16 * LDSSTR + 2 * half];

  float m_run[8], s_run[8];
#pragma unroll
  for (int j = 0; j < 8; ++j) { m_run[j] = -INFINITY; s_run[j] = 0.0f; }

  for (int t = 0; t < 16; ++t) {
    const int col0 = chunk * 2048 + (wave + 8 * t) * 16;
    const float* bBase = cm + (size_t)(col0 + l16) * E_DIM + 2 * half;
    v8f c0 = {}, c1 = {};
#pragma unroll 8
    for (int k = 0; k < E_DIM; k += 8) {       // two independent WMMA chains
      v2f a0 = *(const v2f*)(aBase + k);
      v2f b0 = *(const v2f*)(bBase + k);
      v2f a1 = *(const v2f*)(aBase + k + 4);
      v2f b1 = *(const v2f*)(bBase + k + 4);
      c0 = wmma4(a0, b0, c0);
      c1 = wmma4(a1, b1, c1);
    }
    const float rn = rnorm[col0 + l16] * (1.0f / 0.05f);
#pragma unroll
    for (int j = 0; j < 8; ++j) {
      float x = (c0[j] + c1[j]) * rn;          // logits[row_j][col0+l16]
      float tm = half16_max(x);
      float ts = half16_sum(__expf(x - tm));
      float nm = fmaxf(m_run[j], tm);
      s_run[j] = s_run[j] * __expf(m_run[j] - nm) + ts * __expf(tm - nm);
      m_run[j] = nm;
    }
  }
  if (l16 == 0) {
#pragma unroll
    for (int j = 0; j < 8; ++j) {
      int row = rowBase + j + 8 * half;
      float* p = part + ((size_t)row * 64 + (chunk * 8 + wave)) * 2;
      p[0] = m_run[j]; p[1] = s_run[j];
    }
  }
}

// K4: merge 64 (m,s) partials per row -> logZ; accumulate (logZ - lab_logit)
__global__ __launch_bounds__(256) void rank_merge_kernel(
    const float* __restrict__ part, const float* __restrict__ lablogit,
    float* __restrict__ accs) {
  int wave = threadIdx.x >> 5, lane = threadIdx.x & 31;
  int row = blockIdx.x * 8 + wave;
  const float* p = part + (size_t)row * 64 * 2;
  float m = p[lane * 2 + 0], s = p[lane * 2 + 1];
  {
    float m2 = p[(lane + 32) * 2 + 0], s2 = p[(lane + 32) * 2 + 1];
    float nm = fmaxf(m, m2);
    s = s * __expf(m - nm) + s2 * __expf(m2 - nm);
    m = nm;
  }
#pragma unroll
  for (int d = 16; d >= 1; d >>= 1) {
    float m2 = __shfl_xor(m, d, 32), s2 = __shfl_xor(s, d, 32);
    float nm = fmaxf(m, m2);
    s = s * __expf(m - nm) + s2 * __expf(m2 - nm);
    m = nm;
  }
  if (lane == 0) atomicAdd(&accs[0], (m + __logf(s)) - lablogit[row]);
}

// K5: fused dual-Gram WMMA + online KD partials.
// Per row b track: (ms,ss) for student logZ; (mt,st,wt) with
// wt = sum exp(tx - mt) * (tx - sx), rescaled on max updates.
__global__ __launch_bounds__(256) void kd_stats_kernel(
    const float* __restrict__ batch, const float* __restrict__ teacher,
    const int* __restrict__ labels, float* __restrict__ part) {
  __shared__ float As[16 * LDSSTR];
  __shared__ float At[16 * LDSSTR];
  const int tid = threadIdx.x, wave = tid >> 5, lane = tid & 31;
  const int rowBase = blockIdx.x * 16;
  const int chunk = blockIdx.y;                 // 512 cols per chunk

  for (int i = tid; i < 16 * (E_DIM / 4); i += 256) {
    int r = i / (E_DIM / 4), c4 = (i % (E_DIM / 4)) * 4;
    float4 v = *(const float4*)(batch + (size_t)(rowBase + r) * E_DIM + c4);
    float4 w = *(const float4*)(teacher + (size_t)(rowBase + r) * E_DIM + c4);
    float* ds = &As[r * LDSSTR + c4]; ds[0] = v.x; ds[1] = v.y; ds[2] = v.z; ds[3] = v.w;
    float* dt = &At[r * LDSSTR + c4]; dt[0] = w.x; dt[1] = w.y; dt[2] = w.z; dt[3] = w.w;
  }
  __syncthreads();

  const int half = lane >> 4, l16 = lane & 15;
  const float* aS = &As[l16 * LDSSTR + 2 * half];
  const float* aT = &At[l16 * LDSSTR + 2 * half];

  int lrow[8];
#pragma unroll
  for (int j = 0; j < 8; ++j) lrow[j] = labels[rowBase + j + 8 * half];

  float ms[8], ss[8], mt[8], st[8], wt[8];
#pragma unroll
  for (int j = 0; j < 8; ++j) {
    ms[j] = -INFINITY; ss[j] = 0.0f; mt[j] = -INFINITY; st[j] = 0.0f; wt[j] = 0.0f;
  }

  for (int t = 0; t < 4; ++t) {
    const int col0 = chunk * 512 + (wave + 8 * t) * 16;
    const float* bS = batch + (size_t)(col0 + l16) * E_DIM + 2 * half;
    const float* bT = teacher + (size_t)(col0 + l16) * E_DIM + 2 * half;
    v8f cs = {}, ct = {};
#pragma unroll 8
    for (int k = 0; k < E_DIM; k += 4) {       // two independent WMMA chains
      v2f a = *(const v2f*)(aS + k), b = *(const v2f*)(bS + k);
      v2f a2 = *(const v2f*)(aT + k), b2 = *(const v2f*)(bT + k);
      cs = wmma4(a, b, cs);
      ct = wmma4(a2, b2, ct);
    }
    const int lcol = labels[col0 + l16];
#pragma unroll
    for (int j = 0; j < 8; ++j) {
      float scale = (lrow[j] == lcol) ? (1.0f / 4.0f) : (0.5f / 4.0f); // /TAU
      float xs = cs[j] * scale, xt = ct[j] * scale;
      // student logZ online
      float sm = half16_max(xs);
      float ssum = half16_sum(__expf(xs - sm));
      float nms = fmaxf(ms[j], sm);
      ss[j] = ss[j] * __expf(ms[j] - nms) + ssum * __expf(sm - nms);
      ms[j] = nms;
      // teacher logZ + weighted (tx - sx) sum online
      float tm = half16_max(xt);
      float te = __expf(xt - tm);
      float tsum = half16_sum(te);
      float twsum = half16_sum(te * (xt - xs));
      float nmt = fmaxf(mt[j], tm);
      float e1 = __expf(mt[j] - nmt), e2 = __expf(tm - nmt);
      st[j] = st[j] * e1 + tsum * e2;
      wt[j] = wt[j] * e1 + twsum * e2;
      mt[j] = nmt;
    }
  }
  if (l16 == 0) {
#pragma unroll
    for (int j = 0; j < 8; ++j) {
      int row = rowBase + j + 8 * half;
      float* p = part + ((size_t)row * 32 + (chunk * 8 + wave)) * 5;
      p[0] = ms[j]; p[1] = ss[j]; p[2] = mt[j]; p[3] = st[j]; p[4] = wt[j];
    }
  }
}

// K6: merge 32 KD partials per row; kl = wt/st - logZt + logZs
__global__ __launch_bounds__(256) void kd_merge_kernel(
    const float* __restrict__ part, float* __restrict__ accs) {
  int wave = threadIdx.x >> 5, lane = threadIdx.x & 31;
  int row = blockIdx.x * 8 + wave;
  const float* p = part + ((size_t)row * 32 + lane) * 5;
  float ms = p[0], ss = p[1], mt = p[2], st = p[3], wt = p[4];
#pragma unroll
  for (int d = 16; d >= 1; d >>= 1) {
    float ms2 = __shfl_xor(ms, d, 32), ss2 = __shfl_xor(ss, d, 32);
    float mt2 = __shfl_xor(mt, d, 32), st2 = __shfl_xor(st, d, 32);
    float wt2 = __shfl_xor(wt, d, 32);
    float nms = fmaxf(ms, ms2);
    ss = ss * __expf(ms - nms) + ss2 * __expf(ms2 - nms); ms = nms;
    float nmt = fmaxf(mt, mt2);
    float e1 = __expf(mt - nmt), e2 = __expf(mt2 - nmt);
    st = st * e1 + st2 * e2;
    wt = wt * e1 + wt2 * e2;
    mt = nmt;
  }
  if (lane == 0) {
    float logZs = ms + __logf(ss);
    float logZt = mt + __logf(st);
    atomicAdd(&accs[1], wt / st - logZt + logZs);
  }
}

// K7: scalar combine -> d_out = {loss, loss_rank, loss_kd}
__global__ void finalize_kernel(const float* __restrict__ accs,
                                const int* __restrict__ epoch,
                                float* __restrict__ out) {
  if (threadIdx.x == 0) {
    float lr = accs[0] * (1.0f / (float)B_ROWS);
    float lkd = accs[1] * (1.0f / (float)B_ROWS);
    float ramp = ((float)epoch[0] / 150.0f) * 1.0f * 16.0f;  // alpha*tau^2
    out[0] = lr + ramp * lkd;
    out[1] = lr;
    out[2] = lkd;
  }
}

extern "C" void kernel_launch(void* const* d_in, const int* in_sizes, int n_in,
                              void* d_out, int out_size, void* d_ws, size_t ws_size,
                              hipStream_t stream) {
  const float* batch   = (const float*)d_in[0];   // [2048, 512]
  const float* teacher = (const float*)d_in[1];   // [2048, 512]
  const float* cm      = (const float*)d_in[2];   // [16384, 512]
  const int*   labels  = (const int*)d_in[3];     // [2048]
  const int*   epoch   = (const int*)d_in[4];     // [1]
  float* out = (float*)d_out;                     // {loss, loss_rank, loss_kd}

  // workspace layout (floats), ~2.43 MB total
  float* ws       = (float*)d_ws;
  float* rnorm    = ws;                     // 16384
  float* lablog   = ws + 16384;             // 2048
  float* rankPart = ws + 18432;             // 2048*64*2 = 262144
  float* kdPart   = ws + 18432 + 262144;    // 2048*32*5 = 327680
  float* accs     = ws + 18432 + 262144 + 327680;  // 2

  init_kernel<<<1, 32, 0, stream>>>(accs);
  rnorm_kernel<<<C_DIM / 8, 256, 0, stream>>>(cm, rnorm);
  label_logit_kernel<<<B_ROWS / 8, 256, 0, stream>>>(batch, cm, labels, rnorm, lablog);
  logits_stats_kernel<<<dim3(B_ROWS / 16, 8), 256, 0, stream>>>(batch, cm, rnorm, rankPart);
  rank_merge_kernel<<<B_ROWS / 8, 256, 0, stream>>>(rankPart, lablog, accs);
  kd_stats_kernel<<<dim3(B_ROWS / 16, 4), 256, 0, stream>>>(batch, teacher, labels, kdPart);
  kd_merge_kernel<<<B_ROWS / 8, 256, 0, stream>>>(kdPart, accs);
  finalize_kernel<<<1, 32, 0, stream>>>(accs, epoch, out);
}